// GatedAttention_12867722019339
// MI455X (gfx1250) — compile-verified
//
#include <hip/hip_runtime.h>
#include <hip/hip_bf16.h>
#include <math.h>

typedef __attribute__((ext_vector_type(2))) float v2f;
typedef __attribute__((ext_vector_type(4))) float v4f;
typedef __attribute__((ext_vector_type(8))) float v8f;

#define C_DIM 128
#define KSTEPS 32   // 128 / 4

// ---------------------------------------------------------------------------
// Zero the fp64 global-sum accumulator (must be re-zeroed every launch).
// ---------------------------------------------------------------------------
__global__ void ga_init_kernel(double* gsum) {
    if (threadIdx.x == 0 && blockIdx.x == 0) *gsum = 0.0;
}

// ---------------------------------------------------------------------------
// Pass 1: per-row dual dot product via V_WMMA_F32_16X16X4_F32.
//   D = A(16x4 of x-rows) * B(4x16, col0 = theta1, col1 = theta2, rest 0)
// accumulated over K=128 (32 WMMA steps).
//
// A-layout (ISA 7.12.2, 32-bit A 16x4): lane<16 -> M=lane, K={0,1} in v0,v1;
// lanes 16-31 -> same M, K={2,3}. B mirrors with N on the lane axis.
//
// The B fragment depends only on (kStep, lane), never on the tile, so it is
// pre-masked and staged in LDS once per block: inner loop is then
//   global_load_b64 (A) + ds_load_b64 (B) + v_wmma  -- branch-free, EXEC
// untouched, LOADcnt/DScnt waits fully decoupled.
// ---------------------------------------------------------------------------
__global__ __launch_bounds__(256)
void ga_pass1_kernel(const float* __restrict__ x,
                     const float* __restrict__ th1,
                     const float* __restrict__ th2,
                     const float* __restrict__ Wp,
                     float* __restrict__ atten_u,
                     double* __restrict__ gsum,
                     int BN)
{
    // [kStep][lane] -> v2f B fragment. Lane stride = 2 dwords -> 32 distinct
    // LDS banks per ds_load_b64: conflict-free.
    __shared__ v2f  Blds[KSTEPS * 32];
    __shared__ float sBlock;

    if (threadIdx.x == 0) sBlock = 0.0f;

    // Build masked B fragments (1024 entries, 4 per thread).
    for (int j = threadIdx.x; j < KSTEPS * 32; j += 256) {
        const int ks = j >> 5;
        const int ln = j & 31;
        const int n  = ln & 15;                 // B column
        const int k0 = (ks << 2) + ((ln >> 4) << 1);
        v2f b = {0.0f, 0.0f};
        if (n == 0)      b = *(const v2f*)(th1 + k0);
        else if (n == 1) b = *(const v2f*)(th2 + k0);
        Blds[j] = b;
    }
    __syncthreads();

    const int lane        = threadIdx.x & 31;
    const int waveInBlock = threadIdx.x >> 5;
    const int wavesPerBlk = blockDim.x >> 5;
    const int gwave       = blockIdx.x * wavesPerBlk + waveInBlock;
    const int nwaves      = gridDim.x * wavesPerBlk;
    const int ntiles      = BN >> 4;            // 16 rows per tile

    const int   m  = lane & 15;                 // A row within tile
    const int   kh = (lane >> 4) << 1;          // K sub-offset: 0 or 2
    const float W  = Wp[0];
    const v2f*  Bl = &Blds[lane];               // per-lane B column, step stride 32

    float local = 0.0f;

    for (int tile = gwave; tile < ntiles; tile += nwaves) {
        const int rowBase = tile << 4;
        const float* xrow = x + (size_t)(rowBase + m) * C_DIM + kh;

        v8f acc = {};
        #pragma unroll
        for (int ks = 0; ks < KSTEPS; ++ks) {
            v2f a = *(const v2f*)(xrow + (ks << 2));   // global_load_b64
            v2f b = Bl[ks << 5];                       // ds_load_b64, conflict-free
            acc = __builtin_amdgcn_wmma_f32_16x16x4_f32(
                false, a, false, b, (short)0, acc, false, false);
        }

        // D[m][n]: vgpr r holds rows r (lanes 0-15) and r+8 (lanes 16-31).
        // t1 = column 0 -> lanes 0,16 ; t2 = column 1 -> lanes 1,17.
        float part = 0.0f;
        #pragma unroll
        for (int i = 0; i < 8; ++i) {
            float d0 = acc[i];
            float d1 = __shfl(d0, (lane & 16) | 1, 32); // fetch col-1 value
            if (m == 0) {
                float t1 = tanhf(d0);
                float t2 = 1.0f / (1.0f + expf(-d1));
                float av = expf(W * t1 * t2);
                atten_u[rowBase + ((lane >> 4) << 3) + i] = av;
                part += av;
            }
        }
        local += part;
    }

    // Wave reduce (only lanes 0 and 16 carry data), then block -> fp64 atomic.
    local += __shfl_down(local, 16, 32);
    if (lane == 0) atomicAdd(&sBlock, local);
    __syncthreads();
    if (threadIdx.x == 0) atomicAdd(gsum, (double)sBlock);
}

// ---------------------------------------------------------------------------
// Pass 2: out[row][c] = x[row][c] * atten_u[row] / sum ; atten_out = atten/sum.
// One row per wave per iteration: 32 lanes x float4 = 128 floats (b128 ops).
// ---------------------------------------------------------------------------
__global__ __launch_bounds__(256)
void ga_pass2_kernel(const float* __restrict__ x,
                     const float* __restrict__ atten_u,
                     const double* __restrict__ gsum,
                     float* __restrict__ out,
                     float* __restrict__ atten_out,
                     int BN)
{
    const float inv  = (float)(1.0 / *gsum);
    const int lane   = threadIdx.x & 31;
    const int gwave  = (blockIdx.x * blockDim.x + threadIdx.x) >> 5;
    const int nwaves = (gridDim.x * blockDim.x) >> 5;

    for (int row = gwave; row < BN; row += nwaves) {
        const float a = atten_u[row] * inv;
        const size_t base = (size_t)row * C_DIM + (size_t)lane * 4;
        v4f xv = *(const v4f*)(x + base);              // global_load_b128
        v4f ov = xv * a;
        *(v4f*)(out + base) = ov;                      // global_store_b128
    }

    const int tid = blockIdx.x * blockDim.x + threadIdx.x;
    const int tot = gridDim.x * blockDim.x;
    for (int i = tid; i < BN; i += tot)
        atten_out[i] = atten_u[i] * inv;
}

// ---------------------------------------------------------------------------
extern "C" void kernel_launch(void* const* d_in, const int* in_sizes, int n_in,
                              void* d_out, int out_size, void* d_ws, size_t ws_size,
                              hipStream_t stream) {
    const float* x   = (const float*)d_in[0];
    const float* th1 = (const float*)d_in[1];
    const float* th2 = (const float*)d_in[2];
    const float* W   = (const float*)d_in[3];

    const int C  = 128;
    const int BN = in_sizes[0] / C;                 // 32 * 16384 = 524288

    double* gsum    = (double*)d_ws;                // 8B accumulator
    float*  atten_u = (float*)((char*)d_ws + 16);   // BN floats scratch
    float*  out     = (float*)d_out;
    float*  atten_o = out + (size_t)BN * C;         // second tuple element

    ga_init_kernel<<<1, 1, 0, stream>>>(gsum);
    ga_pass1_kernel<<<1024, 256, 0, stream>>>(x, th1, th2, W, atten_u, gsum, BN);
    ga_pass2_kernel<<<2048, 256, 0, stream>>>(x, atten_u, gsum, out, atten_o, BN);
}